// SelfAttnRNN_3435973837271
// MI455X (gfx1250) — compile-verified
//
#include <hip/hip_runtime.h>
#include <hip/hip_bf16.h>

#define BB 32
#define SS 512
#define HH 128
#define H2 256
#define H3 384
#define H4 512
#define NEG_INF (-1e30f)

typedef float v2f __attribute__((ext_vector_type(2)));
typedef float v8f __attribute__((ext_vector_type(8)));
typedef unsigned int u32x4 __attribute__((ext_vector_type(4)));
typedef int i32x4 __attribute__((ext_vector_type(4)));
typedef int i32x8 __attribute__((ext_vector_type(8)));

// D = A(16x4 f32) * B(4x16 f32) + C(16x16 f32), wave32 WMMA.
__device__ __forceinline__ v8f wmma4(v2f a, v2f b, v8f c) {
  return __builtin_amdgcn_wmma_f32_16x16x4_f32(
      /*neg_a=*/false, a, /*neg_b=*/false, b,
      /*c_mod=*/(short)0, c, /*reuse_a=*/false, /*reuse_b=*/false);
}

__device__ __forceinline__ float sigmoidf_(float x) {
  return 1.0f / (1.0f + __expf(-x));
}

// ---------------------------------------------------------------------------
// K0: transpose weight W[rows][cols] -> WT[cols][rows] so WMMA B-fragment
// loads are lane-contiguous (coalesced) instead of 16-line gathers.
// ---------------------------------------------------------------------------
__global__ void k_tr(const float* __restrict__ src, float* __restrict__ dst,
                     int rows, int cols) {
  int idx = blockIdx.x * 256 + threadIdx.x;
  if (idx < rows * cols) {
    int r = idx / cols, c = idx - r * cols;
    dst[(size_t)c * rows + r] = src[idx];
  }
}

// ---------------------------------------------------------------------------
// K1: keyp[b,s,h] = sum_d vp[b,s,d] * W_vP[h,d]   (32x512)x256 @ 256x128
// Block = 16-row tile (A staged in LDS), 8 waves = 8 N-tiles. Grid 1024.
// ---------------------------------------------------------------------------
__global__ void k_keyp(const float* __restrict__ vp, const float* __restrict__ W_vPT,
                       float* __restrict__ keyp) {
  __shared__ float at[16][260];   // padded: 260 % 64 == 4
  int b = blockIdx.x >> 5, m0 = (blockIdx.x & 31) << 4;
  int tid = threadIdx.x;
  for (int idx = tid; idx < 16 * H2; idx += 256)
    at[idx >> 8][idx & 255] =
        vp[((size_t)(b * SS) + m0 + (idx >> 8)) * H2 + (idx & 255)];
  __syncthreads();
  int lane = tid & 31, wave = tid >> 5;
  int mr = lane & 15, klo = (lane >> 4) << 1;
  int n0 = wave << 4;
  v8f c = {};
  for (int k0 = 0; k0 < H2; k0 += 4) {
    v2f a = *(const v2f*)&at[mr][k0 + klo];
    v2f bv;
    bv.x = W_vPT[(size_t)(k0 + klo) * HH + n0 + mr];
    bv.y = W_vPT[(size_t)(k0 + klo + 1) * HH + n0 + mr];
    c = wmma4(a, bv, c);
  }
  int rbase = m0 + ((lane >> 4) << 3);
  for (int r = 0; r < 8; ++r)
    keyp[((size_t)(b * SS) + rbase + r) * HH + n0 + mr] = c[r];
}

// ---------------------------------------------------------------------------
// K2: q[b,t,h] = sum_d x[b,t,d] * W_vPt[h,d], x = bidir gather of vp.
// Grid 2048 (64 b x 32 m-tiles).
// ---------------------------------------------------------------------------
__global__ void k_q(const float* __restrict__ vp, const int* __restrict__ lens,
                    const float* __restrict__ W_vPtT, float* __restrict__ q) {
  __shared__ float at[16][260];
  int b = blockIdx.x >> 5, m0 = (blockIdx.x & 31) << 4;
  int bb = b & 31;
  int len = lens[bb];
  int tid = threadIdx.x;
  for (int idx = tid; idx < 16 * H2; idx += 256) {
    int row = idx >> 8, col = idx & 255;
    int s = m0 + row;
    int sr = (b < BB) ? s : (s < len ? (len - 1 - s) : s);
    at[row][col] = vp[((size_t)(bb * SS) + sr) * H2 + col];
  }
  __syncthreads();
  int lane = tid & 31, wave = tid >> 5;
  int mr = lane & 15, klo = (lane >> 4) << 1;
  int n0 = wave << 4;
  v8f c = {};
  for (int k0 = 0; k0 < H2; k0 += 4) {
    v2f a = *(const v2f*)&at[mr][k0 + klo];
    v2f bv;
    bv.x = W_vPtT[(size_t)(k0 + klo) * HH + n0 + mr];
    bv.y = W_vPtT[(size_t)(k0 + klo + 1) * HH + n0 + mr];
    c = wmma4(a, bv, c);
  }
  int rbase = m0 + ((lane >> 4) << 3);
  for (int r = 0; r < 8; ++r)
    q[((size_t)(b * SS) + rbase + r) * HH + n0 + mr] = c[r];
}

// ---------------------------------------------------------------------------
// K3: attention for one (b, 16-row t-block); ct GEMM via WMMA, K=512.
// Grid 2048, 256 threads.
// ---------------------------------------------------------------------------
__global__ void k_attn(const float* __restrict__ vp, const int* __restrict__ lens,
                       const float* __restrict__ keyp, const float* __restrict__ q,
                       const float* __restrict__ w_vT, float* __restrict__ vpc) {
  __shared__ float lp[16][516];
  __shared__ float qs[16][128];
  __shared__ float red[16][16];
  __shared__ float rowmax[16];
  __shared__ float rowinv[16];

  int b  = blockIdx.x >> 5;
  int t0 = (blockIdx.x & 31) << 4;
  int bb = b & 31;
  int len = lens[bb];
  int tid = threadIdx.x;
  int wave = tid >> 5, lane = tid & 31;

  for (int idx = tid; idx < 16 * 128; idx += 256)
    qs[idx >> 7][idx & 127] =
        q[((size_t)(b * SS) + t0 + (idx >> 7)) * HH + (idx & 127)];
  __syncthreads();

  // logits: 4 h per lane, wave-reduce; s strided by wave
  float4 w4 = *(const float4*)(w_vT + (lane << 2));
  for (int j = 0; j < 64; ++j) {
    int s = (j << 3) + wave;
    float4 kp = *(const float4*)(keyp + ((size_t)(bb * SS) + s) * HH + (lane << 2));
    bool valid = s < len;
    for (int t = 0; t < 16; ++t) {
      float4 q4 = *(const float4*)&qs[t][lane << 2];
      float p = tanhf(kp.x + q4.x) * w4.x + tanhf(kp.y + q4.y) * w4.y +
                tanhf(kp.z + q4.z) * w4.z + tanhf(kp.w + q4.w) * w4.w;
      p += __shfl_xor(p, 16);
      p += __shfl_xor(p, 8);
      p += __shfl_xor(p, 4);
      p += __shfl_xor(p, 2);
      p += __shfl_xor(p, 1);
      if (lane == 0) lp[t][s] = valid ? p : NEG_INF;
    }
  }
  __syncthreads();

  // softmax stats (normalization folded into the WMMA A fragment)
  {
    int row = tid >> 4, sub = tid & 15;
    float m = -3.4e38f;
    for (int j = 0; j < 32; ++j) m = fmaxf(m, lp[row][(j << 4) + sub]);
    red[row][sub] = m;
    __syncthreads();
    if (tid < 16) {
      float mm = red[tid][0];
      for (int i = 1; i < 16; ++i) mm = fmaxf(mm, red[tid][i]);
      rowmax[tid] = mm;
    }
    __syncthreads();
    float mm = rowmax[row];
    float ss = 0.0f;
    for (int j = 0; j < 32; ++j) {
      int sc = (j << 4) + sub;
      float e = __expf(lp[row][sc] - mm);
      lp[row][sc] = e;
      ss += e;
    }
    red[row][sub] = ss;
    __syncthreads();
    if (tid < 16) {
      float s2 = 0.0f;
      for (int i = 0; i < 16; ++i) s2 += red[tid][i];
      rowinv[tid] = 1.0f / s2;
    }
    __syncthreads();
  }

  // ct = probs(16x512) @ vp[bb](512x256) via WMMA f32 16x16x4
  int mr = lane & 15, klo = (lane >> 4) << 1;
  float inv = rowinv[mr];
  for (int it = 0; it < 2; ++it) {
    int n0 = ((wave << 1) + it) << 4;
    v8f c = {};
    for (int k0 = 0; k0 < SS; k0 += 4) {
      v2f a;
      a.x = lp[mr][k0 + klo] * inv;
      a.y = lp[mr][k0 + klo + 1] * inv;
      const float* vcol = vp + ((size_t)(bb * SS) + k0 + klo) * H2 + n0 + mr;
      v2f bv;
      bv.x = vcol[0];
      bv.y = vcol[H2];
      c = wmma4(a, bv, c);
    }
    int rbase = (lane >> 4) << 3;
    for (int r = 0; r < 8; ++r)
      vpc[((size_t)(b * SS) + t0 + rbase + r) * H4 + H2 + n0 + mr] = c[r];
  }

  // x part of vpc (bidir gather)
  for (int idx = tid; idx < 16 * H2; idx += 256) {
    int tt = idx >> 8, d = idx & 255;
    int s = t0 + tt;
    int sr = (b < BB) ? s : (s < len ? (len - 1 - s) : s);
    vpc[((size_t)(b * SS) + s) * H4 + d] = vp[((size_t)(bb * SS) + sr) * H2 + d];
  }
}

// ---------------------------------------------------------------------------
// K4: xin = sigmoid(vpc @ W_g^T) * vpc, gi = xin @ W_ih^T + b_ih.
// Input tile DMA'd to LDS by the Tensor Data Mover (pad_enable reproduces the
// 516-float bank-conflict row stride: +2 DWORDs every 256 DWORDs).
// Grid 2048, 256 threads.
// ---------------------------------------------------------------------------
__global__ void k_gates(const float* __restrict__ vpc, const float* __restrict__ W_gT,
                        const float* __restrict__ W_ihT, const float* __restrict__ b_ih,
                        float* __restrict__ gi) {
  __shared__ float xa[16 * 516];
  // TDM layout: row r, col c lives at r*516 + c + 2*(c>=256)
#define XA_PTR(r, c) (&xa[(size_t)(r) * 516 + (c) + ((((unsigned)(c)) >> 8) << 1)])
  int b  = blockIdx.x >> 5;
  int t0 = (blockIdx.x & 31) << 4;
  int tid = threadIdx.x;
  const float* tile = vpc + ((size_t)(b * SS) + t0) * H4;

  if (tid == 0) {
    unsigned lds = (unsigned)(unsigned long long)&xa[0];
    unsigned long long ga = (unsigned long long)tile;
    u32x4 g0;
    g0[0] = 1u;                                            // count=1, load-tensor
    g0[1] = lds;                                           // lds_addr
    g0[2] = (unsigned)ga;                                  // global_addr[31:0]
    g0[3] = ((unsigned)(ga >> 32) & 0x1FFFFFFu) | (2u << 30); // addr[56:32], type=2
    i32x8 g1;
    g1[0] = (int)((2u << 16)      // data_size = 4B
                | (1u << 20)      // pad_enable
                | (7u << 22)      // pad_interval = 256 DWORDs
                | (1u << 25));    // pad_amount = 2 DWORDs
    g1[1] = (int)(512u << 16);    // tensor_dim0 = 512
    g1[2] = (int)(16u << 16);     // tensor_dim1 = 16
    g1[3] = (int)(512u << 16);    // tile_dim0 = 512
    g1[4] = 16;                   // tile_dim1 = 16
    g1[5] = 512;                  // tensor_dim0_stride = 512
    g1[6] = 0;
    g1[7] = 0;
    i32x4 gz = {};
#if __clang_major__ >= 23
    i32x8 gz8 = {};
    __builtin_amdgcn_tensor_load_to_lds(g0, g1, gz, gz, gz8, 0);
#else
    __builtin_amdgcn_tensor_load_to_lds(g0, g1, gz, gz, 0);
#endif
    __builtin_amdgcn_s_wait_tensorcnt(0);
  }
  __syncthreads();

  int lane = tid & 31, wave = tid >> 5;
  int mr = lane & 15, klo = (lane >> 4) << 1;
  int rbase = (lane >> 4) << 3;

  // Phase A: gate + elementwise product; results held in registers.
  float xg[4][8];
  for (int i = 0; i < 4; ++i) {
    int n0 = (wave + 8 * i) << 4;
    v8f c = {};
    for (int k0 = 0; k0 < H4; k0 += 4) {
      v2f a = *(const v2f*)XA_PTR(mr, k0 + klo);
      v2f bv;
      bv.x = W_gT[(size_t)(k0 + klo) * H4 + n0 + mr];
      bv.y = W_gT[(size_t)(k0 + klo + 1) * H4 + n0 + mr];
      c = wmma4(a, bv, c);
    }
    for (int r = 0; r < 8; ++r)
      xg[i][r] = sigmoidf_(c[r]) * (*XA_PTR(rbase + r, n0 + mr));
  }
  __syncthreads();
  for (int i = 0; i < 4; ++i) {
    int n0 = (wave + 8 * i) << 4;
    for (int r = 0; r < 8; ++r) *XA_PTR(rbase + r, n0 + mr) = xg[i][r];
  }
  __syncthreads();

  // Phase B: gi = xin @ W_ih^T + b_ih
  for (int i = 0; i < 3; ++i) {
    int n0 = (wave + 8 * i) << 4;
    v8f c = {};
    for (int k0 = 0; k0 < H4; k0 += 4) {
      v2f a = *(const v2f*)XA_PTR(mr, k0 + klo);
      v2f bv;
      bv.x = W_ihT[(size_t)(k0 + klo) * H3 + n0 + mr];
      bv.y = W_ihT[(size_t)(k0 + klo + 1) * H3 + n0 + mr];
      c = wmma4(a, bv, c);
    }
    for (int r = 0; r < 8; ++r) {
      int row = rbase + r, col = n0 + mr;
      gi[((size_t)(b * SS) + t0 + row) * H3 + col] = c[r] + b_ih[col];
    }
  }
#undef XA_PTR
}

// ---------------------------------------------------------------------------
// K5: sequential GRU scan. 64 WGs (one per sequence), 128 threads.
// ---------------------------------------------------------------------------
__global__ void k_scan(const float* __restrict__ gi, const float* __restrict__ W_hh,
                       const float* __restrict__ b_hh, const int* __restrict__ lens,
                       float* __restrict__ out) {
  __shared__ float hs[128];
  int b = blockIdx.x;
  int bb = b & 31;
  int half = (b >> 5) & 1;
  int len = lens[bb];
  int tid = threadIdx.x;
  hs[tid] = 0.0f;
  __syncthreads();
  float bh0 = b_hh[tid], bh1 = b_hh[tid + 128], bh2 = b_hh[tid + 256];
  const float* w0 = W_hh + (size_t)tid * 128;
  const float* w1 = W_hh + (size_t)(tid + 128) * 128;
  const float* w2 = W_hh + (size_t)(tid + 256) * 128;
  for (int t = 0; t < SS; ++t) {
    const float* gir = gi + ((size_t)(b * SS) + t) * H3;
    if (t + 1 < SS) __builtin_prefetch(gir + H3, 0, 1);   // global_prefetch_b8
    float g0 = bh0, g1 = bh1, g2 = bh2;
    for (int k = 0; k < 128; k += 4) {
      float4 hv = *(const float4*)&hs[k];
      float4 a0 = *(const float4*)(w0 + k);
      float4 a1 = *(const float4*)(w1 + k);
      float4 a2 = *(const float4*)(w2 + k);
      g0 += hv.x * a0.x + hv.y * a0.y + hv.z * a0.z + hv.w * a0.w;
      g1 += hv.x * a1.x + hv.y * a1.y + hv.z * a1.z + hv.w * a1.w;
      g2 += hv.x * a2.x + hv.y * a2.y + hv.z * a2.z + hv.w * a2.w;
    }
    float r = sigmoidf_(gir[tid] + g0);
    float z = sigmoidf_(gir[tid + 128] + g1);
    float n = tanhf(gir[tid + 256] + r * g2);
    float hold = hs[tid];
    float hnew = (1.0f - z) * n + z * hold;
    if (t >= len) hnew = 0.0f;
    __syncthreads();
    hs[tid] = hnew;
    __syncthreads();
    out[((size_t)(bb * SS) + t) * H2 + half * HH + tid] = hnew;
  }
}

// ---------------------------------------------------------------------------
extern "C" void kernel_launch(void* const* d_in, const int* in_sizes, int n_in,
                              void* d_out, int out_size, void* d_ws, size_t ws_size,
                              hipStream_t stream) {
  (void)in_sizes; (void)n_in; (void)out_size; (void)ws_size;
  const float* vp    = (const float*)d_in[0];
  const int*   lens  = (const int*)d_in[1];
  const float* W_vP  = (const float*)d_in[2];
  const float* W_vPt = (const float*)d_in[3];
  const float* w_vT  = (const float*)d_in[4];
  const float* W_g   = (const float*)d_in[5];
  const float* W_ih  = (const float*)d_in[6];
  const float* W_hh  = (const float*)d_in[7];
  const float* b_ih  = (const float*)d_in[8];
  const float* b_hh  = (const float*)d_in[9];
  float* out = (float*)d_out;

  float* ws    = (float*)d_ws;
  float* keyp  = ws;                                  // 32*512*128
  float* q     = keyp  + (size_t)BB * SS * HH;        // 64*512*128
  float* vpc   = q     + (size_t)2 * BB * SS * HH;    // 64*512*512
  float* gi    = vpc   + (size_t)2 * BB * SS * H4;    // 64*512*384
  float* W_gT  = gi    + (size_t)2 * BB * SS * H3;    // 512*512
  float* W_ihT = W_gT  + (size_t)H4 * H4;             // 512*384
  float* W_vPT = W_ihT + (size_t)H4 * H3;             // 256*128
  float* W_vPtT= W_vPT + (size_t)H2 * HH;             // 256*128

  k_tr<<<(H4 * H4 + 255) / 256, 256, 0, stream>>>(W_g, W_gT, H4, H4);
  k_tr<<<(H3 * H4 + 255) / 256, 256, 0, stream>>>(W_ih, W_ihT, H3, H4);
  k_tr<<<(HH * H2 + 255) / 256, 256, 0, stream>>>(W_vP, W_vPT, HH, H2);
  k_tr<<<(HH * H2 + 255) / 256, 256, 0, stream>>>(W_vPt, W_vPtT, HH, H2);

  k_keyp <<<1024, 256, 0, stream>>>(vp, W_vPT, keyp);
  k_q    <<<2048, 256, 0, stream>>>(vp, lens, W_vPtT, q);
  k_attn <<<2048, 256, 0, stream>>>(vp, lens, keyp, q, w_vT, vpc);
  k_gates<<<2048, 256, 0, stream>>>(vpc, W_gT, W_ihT, b_ih, gi);
  k_scan <<<64, 128, 0, stream>>>(gi, W_hh, b_hh, lens, out);
}